// Signature_4200478016023
// MI455X (gfx1250) — compile-verified
//
#include <hip/hip_runtime.h>
#include <hip/hip_bf16.h>

// Truncated path signature depth=3, B=512, L=512, C=8, fp32.
// Restructured as: channel prefix sums -> per-chunk level-2 cumsum -> one
// (64 x 511) @ (511 x 8) fp32 GEMM per batch done with V_WMMA_F32_16X16X4_F32.

typedef __attribute__((ext_vector_type(2))) float v2f;
typedef __attribute__((ext_vector_type(8))) float v8f;

#define NB    512   // batches
#define LL    512   // stream length
#define NC    8     // channels
#define LM1   511   // increments
#define DXSTR 9     // padded LDS row stride for [t][c]
#define MSTR  65    // padded LDS row stride for [t][ij]
#define OUTB  584   // 8 + 64 + 512 outputs per batch

__global__ __launch_bounds__(256) void sig_depth3_kernel(
    const float* __restrict__ path, float* __restrict__ out)
{
    __shared__ float s_dx[LL * DXSTR];    // dx[t][c], t=511 zero pad
    __shared__ float s_P [LL * DXSTR];    // exclusive prefix sums of dx
    __shared__ float s_M [64 * MSTR];     // per-chunk A matrix M[tloc][ij]
    __shared__ float s_A2[64];            // running level-2 signature
    __shared__ float s_seg[8][8];         // scan segment sums [c][seg]
    __shared__ float s_stage[4][32][8];   // K-half reduction staging

    const int tid = threadIdx.x;
    const int b   = blockIdx.x;
    const float* pb = path + (size_t)b * LL * NC;

    // ---- Phase 1: increments into LDS (coalesced global reads) ----
    for (int idx = tid; idx < LL * NC; idx += 256) {
        int t = idx >> 3, c = idx & 7;
        float d = 0.f;
        if (t < LM1) d = pb[(t + 1) * NC + c] - pb[t * NC + c];
        s_dx[t * DXSTR + c] = d;
    }
    if (tid < 64) s_A2[tid] = 0.f;
    __syncthreads();

    // ---- Phase 2: exclusive prefix sums per channel (3-phase blocked scan) ----
    if (tid < 64) {                      // 8 channels x 8 segments of 64
        int c = tid & 7, seg = tid >> 3, base = seg * 64;
        float s = 0.f;
        for (int t = 0; t < 64; ++t) s += s_dx[(base + t) * DXSTR + c];
        s_seg[c][seg] = s;
    }
    __syncthreads();
    if (tid < 8) {
        float run = 0.f;
        for (int seg = 0; seg < 8; ++seg) {
            float v = s_seg[tid][seg]; s_seg[tid][seg] = run; run += v;
        }
    }
    __syncthreads();
    if (tid < 64) {
        int c = tid & 7, seg = tid >> 3, base = seg * 64;
        float run = s_seg[c][seg];
        for (int t = 0; t < 64; ++t) {
            s_P[(base + t) * DXSTR + c] = run;
            run += s_dx[(base + t) * DXSTR + c];
        }
    }

    // ---- Phase 3: chunked level-2 cumsum + level-3 GEMM via WMMA ----
    const int wave  = tid >> 5;          // 8 wave32s
    const int lane  = tid & 31;
    const int mtile = wave & 3;          // 16-row tile of the 64 (ij) rows
    const int khalf = wave >> 2;         // 0: k 0..31 of chunk, 1: 32..63
    const int lrow  = lane & 15;         // A-matrix row within tile
    const int koffl = (lane < 16) ? 0 : 2; // ISA 16x4 f32 A/B lane->K mapping
    const int ncol  = lane & 15;         // B/D column (valid < 8)

    v8f acc = {};

    for (int q = 0; q < 8; ++q) {
        __syncthreads();                 // s_M free (prev chunk WMMAs done)
        if (tid < 64) {
            // M[t][ij] = A2prev + 0.5*P_i*dx_j + (1/6)*dx_i*dx_j ; G cumsum
            int ij = tid, i = ij >> 3, j = ij & 7;
            float base = s_A2[ij], run = 0.f;
            int t0 = q * 64;
            for (int tl = 0; tl < 64; ++tl) {
                float pi = s_P [(t0 + tl) * DXSTR + i];
                float dj = s_dx[(t0 + tl) * DXSTR + j];
                float di = s_dx[(t0 + tl) * DXSTR + i];
                float pidj = pi * dj;
                float didj = di * dj;
                s_M[tl * MSTR + ij] = base + run + 0.5f * pidj
                                      + (1.0f / 6.0f) * didj;
                run += pidj + 0.5f * didj;   // G[t]_{ij}
            }
            s_A2[ij] = base + run;
        }
        __syncthreads();

        // S3 tile GEMM: A = M (64 x 64), B = dx (64 x 8 padded to 16), K by 4
        const int t0 = q * 64;
        const int kb = khalf * 32;
        for (int ks = 0; ks < 8; ++ks) {
            int kl = kb + ks * 4;
            v2f a, bf;
            a.x = s_M[(kl + koffl    ) * MSTR + mtile * 16 + lrow];
            a.y = s_M[(kl + koffl + 1) * MSTR + mtile * 16 + lrow];
            float bx = 0.f, by = 0.f;
            if (ncol < 8) {
                bx = s_dx[(t0 + kl + koffl    ) * DXSTR + ncol];
                by = s_dx[(t0 + kl + koffl + 1) * DXSTR + ncol];
            }
            bf.x = bx; bf.y = by;
            acc = __builtin_amdgcn_wmma_f32_16x16x4_f32(
                false, a, false, bf, (short)0, acc, false, false);
        }
    }

    __syncthreads();
    // ---- Phase 4: reduce K-halves, store outputs ----
    if (khalf == 1) {
        for (int v = 0; v < 8; ++v) s_stage[mtile][lane][v] = acc[v];
    }
    __syncthreads();

    float* ob = out + (size_t)b * OUTB;
    if (khalf == 0) {
        if (ncol < 8) {
            int mofs = (lane < 16) ? 0 : 8;   // ISA 16x16 f32 D layout
            for (int v = 0; v < 8; ++v) {
                float val  = acc[v] + s_stage[mtile][lane][v];
                int   grow = mtile * 16 + mofs + v;      // flattened (i*8+j)
                ob[72 + grow * 8 + ncol] = val;          // S3
            }
        }
    }
    if (tid < 64) ob[8 + tid] = s_A2[tid];               // S2 == final A2
    if (tid < 8)  ob[tid]     = s_P[LM1 * DXSTR + tid];  // S1 == total sum
}

extern "C" void kernel_launch(void* const* d_in, const int* in_sizes, int n_in,
                              void* d_out, int out_size, void* d_ws, size_t ws_size,
                              hipStream_t stream)
{
    (void)in_sizes; (void)n_in; (void)d_ws; (void)ws_size; (void)out_size;
    const float* path = (const float*)d_in[0];
    float* out = (float*)d_out;
    sig_depth3_kernel<<<NB, 256, 0, stream>>>(path, out);
}